// mbLoss_2405181685891
// MI455X (gfx1250) — compile-verified
//
#include <hip/hip_runtime.h>
#include <hip/hip_bf16.h>
#include <math.h>

typedef __attribute__((ext_vector_type(2))) float v2f;
typedef __attribute__((ext_vector_type(8))) float v8f;

namespace {
constexpr int kGXc = 128;
constexpr int kGYc = 128;
constexpr int kAc  = 9;
constexpr int kNCell = 32 * kGXc * kGYc;          // 524288 grid cells
constexpr int kNElem = kNCell * kAc;              // 4718592 anchor elements
constexpr int kBlocks  = 1024;
constexpr int kThreads = 256;
}

// Exact f32 wave reduction on the matrix pipe:
//   D = A(16x4) x ones(4x16) + C  accumulates row sums (all 16 columns equal).
// C/D layout: lanes 0..15 hold rows 0..7 in v[0..7], lanes 16..31 hold rows
// 8..15, so summing the 8 accumulator regs per lane and adding lane0+lane16
// yields the total of 64*nchunks values. Must run with a full wave (EXEC=~0).
__device__ __forceinline__ float wave_wmma_sum(const float* src, int nchunks, int lane) {
  v2f ones; ones[0] = 1.0f; ones[1] = 1.0f;
  v8f acc = {};
  for (int c = 0; c < nchunks; ++c) {
    v2f a;
    a[0] = src[c * 64 + lane * 2 + 0];
    a[1] = src[c * 64 + lane * 2 + 1];
    // 8 args: (neg_a, A, neg_b, B, c_mod, C, reuse_a, reuse_b)
    acc = __builtin_amdgcn_wmma_f32_16x16x4_f32(false, a, false, ones,
                                                (short)0, acc, false, false);
  }
  float s = ((acc[0] + acc[1]) + (acc[2] + acc[3])) +
            ((acc[4] + acc[5]) + (acc[6] + acc[7]));
  return __shfl(s, 0, 32) + __shfl(s, 16, 32);
}

__global__ __launch_bounds__(kThreads) void ciou_partial(
    const float* __restrict__ outp, const float* __restrict__ tgt,
    float* __restrict__ part) {
  const float invGX = 1.0f / kGXc;
  const float invGY = 1.0f / kGYc;

  float s_all = 0.f, s_iou = 0.f, s_huh = 0.f, s_zsd = 0.f, s_cnt = 0.f;

  for (int i = blockIdx.x * kThreads + threadIdx.x; i < kNElem;
       i += kBlocks * kThreads) {
    const int cell = i / kAc;                       // mul+shift
    const int gy = cell & (kGYc - 1);
    const int gx = (cell >> 7) & (kGXc - 1);

    const float* st = tgt + (size_t)cell * 18;      // target[...,0,:] (Q=3,C=6)
    const float acx = st[0], acy = st[1], aw = st[2], ah = st[3];
    const bool pos = st[5] > 0.8f;

    const float* o = outp + (size_t)i * 5;
    const float o0 = o[0], o1 = o[1], o2 = o[2], o3 = o[3], obj = o[4];

    const float zx = ((float)gx + 0.5f) * invGX;
    const float zy = ((float)gy + 0.5f) * invGY;
    const float bcx = o0 - 0.5f + zx;
    const float bcy = o1 - 0.5f + zy;
    const float bw  = o2 - 0.5f + invGX;
    const float bh  = o3 - 0.5f + invGY;

    const float ax1 = acx - 0.5f * aw, ax2 = acx + 0.5f * aw;
    const float ay1 = acy - 0.5f * ah, ay2 = acy + 0.5f * ah;
    const float bx1 = bcx - 0.5f * bw, bx2 = bcx + 0.5f * bw;
    const float by1 = bcy - 0.5f * bh, by2 = bcy + 0.5f * bh;

    const float iw = fminf(ax2, bx2) - fmaxf(ax1, bx1);
    const float ih = fminf(ay2, by2) - fmaxf(ay1, by1);
    const float cross = (iw > 0.f && ih > 0.f) ? iw * ih : 0.f;
    const float uni = aw * ah + bw * bh - cross;    // area_a + area_b - cross
    const float iou = cross / (uni + 1e-6f);

    const float dx = bcx - acx, dy = bcy - acy;     // box centers directly
    const float d2 = dx * dx + dy * dy;             // only d^2 is ever used
    const float cw  = fmaxf(ax2, bx2) - fminf(ax1, bx1);
    const float chh = fmaxf(ay2, by2) - fminf(ay1, by1);
    const float c2 = cw * cw + chh * chh;           // only c^2 is ever used
    const float diou = iou - d2 / c2;

    const float dat = atanf(aw / ah) - atanf(bw / bh);
    const float v = 0.4052847345693511f * dat * dat; // 4/pi^2
    const float alpha = v / (1.f - iou + v);
    const float los = 1.f - (diou - alpha * v);

    if (pos) {
      const float om = 1.f - obj;
      s_all += los + om;
      s_iou += iou;
      s_huh += om;
      s_zsd += om * om;
      s_cnt += 1.f;                                  // per-anchor => jsq directly
    } else {
      s_zsd += obj * obj;
    }
  }

  __shared__ float lds[5 * kThreads];
  const int t = threadIdx.x;
  lds[0 * kThreads + t] = s_all;
  lds[1 * kThreads + t] = s_iou;
  lds[2 * kThreads + t] = s_huh;
  lds[3 * kThreads + t] = s_zsd;
  lds[4 * kThreads + t] = s_cnt;
  __syncthreads();

  if (t < 32) {                                      // wave 0, EXEC all ones
    for (int j = 0; j < 5; ++j) {
      const float tot = wave_wmma_sum(&lds[j * kThreads], kThreads / 64, t);
      if (t == 0) part[j * kBlocks + blockIdx.x] = tot;
    }
  }
}

__global__ void ciou_final(const float* __restrict__ part,
                           float* __restrict__ outv) {
  const int t = threadIdx.x;                         // launched with 32 threads
  float tot[5];
  for (int j = 0; j < 5; ++j)
    tot[j] = wave_wmma_sum(&part[j * kBlocks], kBlocks / 64, t);
  if (t == 0) {
    const float jsq = tot[4];                        // sum(pos) * A
    const float nel = (float)kNElem;                 // jsq + qit == N always
    outv[0] = tot[0] / jsq + tot[3] / nel;
    outv[1] = tot[1] / jsq;
    outv[2] = tot[2] / jsq;
  }
}

extern "C" void kernel_launch(void* const* d_in, const int* /*in_sizes*/, int /*n_in*/,
                              void* d_out, int /*out_size*/, void* d_ws, size_t /*ws_size*/,
                              hipStream_t stream) {
  const float* outp = (const float*)d_in[0];
  const float* tgt  = (const float*)d_in[1];
  float* part = (float*)d_ws;                        // 5 * 1024 floats = 20 KB
  ciou_partial<<<kBlocks, kThreads, 0, stream>>>(outp, tgt, part);
  ciou_final<<<1, 32, 0, stream>>>(part, (float*)d_out);
}